// SaltMHA_71914932404644
// MI455X (gfx1250) — compile-verified
//
#include <hip/hip_runtime.h>

typedef __attribute__((ext_vector_type(2))) float v2f;
typedef __attribute__((ext_vector_type(8))) float v8f;

#define B_ 4
#define T_ 1024
#define E_ 512
#define H_ 8
#define D_ 64

// ---------------------------------------------------------------------------
// Kernel 1: sequential tanh recurrence -> CC buffer (B,H,T,D), scaled by 1/8
// ---------------------------------------------------------------------------
__global__ void cc_scan_kernel(const float* __restrict__ target,
                               const float* __restrict__ encoder,
                               float* __restrict__ cc_out) {
    const int bh = blockIdx.x;          // 0..31
    const int b  = bh >> 3;
    const int h  = bh & 7;
    const int d  = threadIdx.x;         // 0..63

    const float pe = encoder[h * D_ + d];
    const float* kbase = target + (size_t)(b * T_) * E_ + h * D_ + d;
    float*       obase = cc_out + (size_t)(bh * T_) * D_ + d;

    float carry = kbase[0];             // cc0 = k[:,:,0,:]
    for (int t = 0; t < T_; ++t) {
        float cc = tanhf(carry * pe);
        obase[t * D_] = cc * 0.125f;    // fold 1/sqrt(64)
        float knext = (t + 1 < T_) ? kbase[(size_t)(t + 1) * E_] : 0.0f;
        carry = knext * cc;
    }
}

// ---------------------------------------------------------------------------
// Kernel 2: per (b,h, 16-row q tile):
//   S = Q*K^T and G = Q*CC^T via v_wmma_f32_16x16x4_f32, row softmax of G in
//   LDS, fused epilogue (alpha*S + beta*softmax)/gamma.
// 256 threads = 8 waves; wave w owns columns [w*128, w*128+128).
// All per-tile deltas are compile-time constants folded into load/store
// immediate offsets from a single hoisted base pointer per operand.
// ---------------------------------------------------------------------------
__global__ void __launch_bounds__(256)
mha_fused_kernel(const float* __restrict__ source,
                 const float* __restrict__ target,
                 const float* __restrict__ ccbuf,
                 const float* __restrict__ alpha_p,
                 const float* __restrict__ beta_p,
                 const float* __restrict__ gamma_p,
                 float* __restrict__ out) {
    __shared__ float sG[16 * T_];       // 64 KB grammar row-block
    __shared__ float sRed[16][16];      // per-row partial reductions
    __shared__ float sRowMax[16];
    __shared__ float sRowSum[16];

    const int bh = blockIdx.x >> 6;     // 0..31
    const int qt = blockIdx.x & 63;     // 0..63 (16-row tiles)
    const int b  = bh >> 3;
    const int h  = bh & 7;

    const int tid   = threadIdx.x;
    const int wave  = tid >> 5;         // 0..7
    const int lane  = tid & 31;
    const int lrow  = lane & 15;        // row-in-tile for A, col-in-tile for B
    const int khalf = lane >> 4;        // 0 or 1
    const int kb    = khalf * 2;        // K-pair base within each group of 4

    // ---- hoisted base pointers (all j/k4 deltas become immediate offsets)
    const float* qbase = source +
        (size_t)((b * T_ + qt * 16 + lrow) * E_ + h * D_ + kb);
    const float* kbase = target +
        (size_t)((b * T_ + wave * 128 + lrow) * E_ + h * D_ + kb);
    const float* cbase = ccbuf +
        (size_t)((bh * T_ + wave * 128 + lrow) * D_ + kb);

    // ---- load Q fragment: lane holds Q[qrow][k4*4 + kb + {0,1}], 16 k-steps
    v2f qf[16];
#pragma unroll
    for (int k4 = 0; k4 < 16; ++k4)
        qf[k4] = *(const v2f*)(qbase + k4 * 4);

    v8f saccs[8];

    // ---- phase 1: dual GEMM over this wave's 8 column tiles
#pragma unroll
    for (int j = 0; j < 8; ++j) {
        v8f sacc = {};
        v8f gacc = {};
#pragma unroll
        for (int k4 = 0; k4 < 16; ++k4) {
            v2f kf = *(const v2f*)(kbase + j * 16 * E_ + k4 * 4);
            v2f cf = *(const v2f*)(cbase + j * 16 * D_ + k4 * 4);
            sacc = __builtin_amdgcn_wmma_f32_16x16x4_f32(
                false, qf[k4], false, kf, (short)0, sacc, false, false);
            gacc = __builtin_amdgcn_wmma_f32_16x16x4_f32(
                false, qf[k4], false, cf, (short)0, gacc, false, false);
        }
        saccs[j] = sacc;

        // D layout: VGPR v, lanes0-15 -> row v, lanes16-31 -> row v+8
        float* gdst = &sG[(8 * khalf) * T_ + wave * 128 + j * 16 + lrow];
#pragma unroll
        for (int v = 0; v < 8; ++v)
            gdst[v * T_] = gacc[v];
    }
    __syncthreads();

    // ---- phase 2: row softmax (16 threads/row, stride-16 = bank clean)
    const int srow = tid >> 4;          // 0..15
    const int sseg = tid & 15;          // 0..15
    {
        float* gp = &sG[srow * T_];
        float m = -3.402823466e38f;
#pragma unroll 8
        for (int i = 0; i < 64; ++i)
            m = fmaxf(m, gp[sseg + 16 * i]);
        sRed[srow][sseg] = m;
    }
    __syncthreads();
    if (tid < 16) {
        float mm = sRed[tid][0];
#pragma unroll
        for (int i = 1; i < 16; ++i) mm = fmaxf(mm, sRed[tid][i]);
        sRowMax[tid] = mm;
    }
    __syncthreads();
    {
        float rmax = sRowMax[srow];
        float* gp = &sG[srow * T_];
        float s = 0.0f;
#pragma unroll 8
        for (int i = 0; i < 64; ++i) {
            float e = __expf(gp[sseg + 16 * i] - rmax);
            gp[sseg + 16 * i] = e;      // overwrite with exp values
            s += e;
        }
        sRed[srow][sseg] = s;
    }
    __syncthreads();
    if (tid < 16) {
        float ss = 0.0f;
#pragma unroll
        for (int i = 0; i < 16; ++i) ss += sRed[tid][i];
        sRowSum[tid] = ss;
    }
    __syncthreads();

    // ---- phase 3: fused epilogue, one base pointer + immediate offsets
    const float a      = alpha_p[0];
    const float be     = beta_p[0];
    const float inv_ga = 1.0f / gamma_p[0];

    float* obase = out +
        ((size_t)(bh * T_) + qt * 16 + 8 * khalf) * T_ + wave * 128 + lrow;
    const float* gsrc = &sG[(8 * khalf) * T_ + wave * 128 + lrow];

#pragma unroll
    for (int j = 0; j < 8; ++j) {
        v8f sacc = saccs[j];
#pragma unroll
        for (int v = 0; v < 8; ++v) {
            float sm = gsrc[v * T_ + j * 16] / sRowSum[v + 8 * khalf];
            float val = (a * sacc[v] + be * sm) * inv_ga;
            obase[(size_t)v * T_ + j * 16] = val;
        }
    }
}

// ---------------------------------------------------------------------------
extern "C" void kernel_launch(void* const* d_in, const int* in_sizes, int n_in,
                              void* d_out, int out_size, void* d_ws, size_t ws_size,
                              hipStream_t stream) {
    const float* source  = (const float*)d_in[0];   // (B,T,E)
    const float* target  = (const float*)d_in[1];   // (B,T,E)
    const float* encoder = (const float*)d_in[2];   // (1,H,1,D)
    const float* alpha   = (const float*)d_in[3];
    const float* beta    = (const float*)d_in[4];
    const float* gamma   = (const float*)d_in[5];
    float* out = (float*)d_out;                     // (B, H*T, T)
    float* cc  = (float*)d_ws;                      // B*H*T*D f32 = 8 MB

    cc_scan_kernel<<<dim3(B_ * H_), dim3(D_), 0, stream>>>(target, encoder, cc);
    mha_fused_kernel<<<dim3(B_ * H_ * (T_ / 16)), dim3(256), 0, stream>>>(
        source, target, cc, alpha, beta, gamma, out);
}